// ReconstructionLoss_4698694221980
// MI455X (gfx1250) — compile-verified
//
#include <hip/hip_runtime.h>
#include <hip/hip_bf16.h>

// ---------------------------------------------------------------------------
// Chamfer loss on MI455X (gfx1250, wave32).
// Per 16x16 distance tile, one V_WMMA_F32_16X16X4_F32 with C = inline 0:
//   A row i = [-2x0, -2x1, -2x2, 1]    (built once per wave, -2 baked in)
//   B col j = [ y0,   y1,   y2, |y|^2] (pre-swizzled to exact B-VGPR layout)
//   A*B = -2 x.y + |y_j|^2
// |x_i|^2 is row-constant -> added once per strip after the row-min reduce.
// ---------------------------------------------------------------------------

typedef __attribute__((ext_vector_type(2))) float v2f;
typedef __attribute__((ext_vector_type(8))) float v8f;

#define CH_B 16
#define CH_P 4096
#define CH_Q 4096

// raw:  float4 {c0,c1,c2,|c|^2} per point (A-operand + row-norm source)
// blay: per 16-point tile, 32 x float2 in B-operand lane order:
//       [tile*32 + m]      = {c0, c1}      (lanes 0-15:  K0,K1)
//       [tile*32 + 16 + m] = {c2, |c|^2}   (lanes 16-31: K2,K3)
__global__ void prep_gen_kernel(const float* __restrict__ pg,
                                float4* __restrict__ raw,
                                float2* __restrict__ blay, int n) {
    int i = blockIdx.x * blockDim.x + threadIdx.x;
    if (i < n) {
        float x = pg[3 * i + 0];
        float y = pg[3 * i + 1];
        float z = pg[3 * i + 2];
        float nn = x * x + y * y + z * z;
        raw[i] = make_float4(x, y, z, nn);
        int t = i >> 4, m = i & 15;
        blay[t * 32 + m]      = make_float2(x, y);
        blay[t * 32 + 16 + m] = make_float2(z, nn);
    }
}

__global__ void prep_gt_kernel(const float* __restrict__ label,
                               float4* __restrict__ raw,
                               float2* __restrict__ blay, int n) {
    int j = blockIdx.x * blockDim.x + threadIdx.x;
    if (j < n) {
        const float inv = 1.0f / 128.0f;
        float x = (label[5 * j + 1] - 128.0f) * inv;
        float y = (label[5 * j + 2] - 128.0f) * inv;
        float z = (label[5 * j + 3] - 128.0f) * inv;
        float nn = x * x + y * y + z * z;
        raw[j] = make_float4(x, y, z, nn);
        int t = j >> 4, m = j & 15;
        blay[t * 32 + m]      = make_float2(x, y);
        blay[t * 32 + 16 + m] = make_float2(z, nn);
    }
}

__global__ void zero_kernel(float* __restrict__ p, int n) {
    int i = blockIdx.x * blockDim.x + threadIdx.x;
    if (i < n) p[i] = 0.0f;
}

// One wave owns a 16-row strip of one batch; scans all column tiles.
__global__ void __launch_bounds__(256)
chamfer_rowmin_kernel(const float4* __restrict__ rows,
                      const float2* __restrict__ colsB,
                      float* __restrict__ accum,
                      int NR, int NC) {
    const int lane          = threadIdx.x & 31;
    const int waveInBlock   = threadIdx.x >> 5;
    const int wavesPerBlock = blockDim.x >> 5;
    const int w             = blockIdx.x * wavesPerBlock + waveInBlock;

    const int stripsPerBatch = NR >> 4;
    const int batch = w / stripsPerBatch;
    const int strip = w % stripsPerBatch;

    const int  m  = lane & 15;
    const bool hi = lane >= 16;

    // A operand (once per wave): lanes 0-15 K0/K1 = -2x0/-2x1,
    //                            lanes 16-31 K2/K3 = -2x2/1.
    const float4 xa = rows[batch * NR + strip * 16 + m];
    v2f a;
    a.x = -2.0f * (hi ? xa.z : xa.x);
    a.y = hi ? 1.0f : (-2.0f * xa.y);

    const v8f czero = {0.f, 0.f, 0.f, 0.f, 0.f, 0.f, 0.f, 0.f};

    v8f rmin;
#pragma unroll
    for (int k = 0; k < 8; ++k) rmin[k] = 3.0e38f;

    // Per-lane column stream: lane L of tile t lives at cb + t*32 + L.
    const float2* cb = colsB + (size_t)batch * NC * 2 + lane;
    const int ntiles = NC >> 4;

#pragma unroll 4
    for (int t = 0; t < ntiles; ++t) {
        const float2* ldaddr = cb + t * 32;
        const float2  bv     = *ldaddr;
        // Prefetch 8 tiles (2048 B) ahead via immediate offset from the same
        // base register -> folds into global_prefetch_b8 ioffset, no VALU.
        __builtin_prefetch((const char*)ldaddr + 8 * 32 * sizeof(float2), 0, 1);

        v2f b;
        b.x = bv.x;
        b.y = bv.y;

        // (neg_a, A, neg_b, B, c_mod, C, reuse_a, reuse_b); C = inline 0
        v8f d = __builtin_amdgcn_wmma_f32_16x16x4_f32(
            false, a, false, b, (short)0, czero, false, false);

#pragma unroll
        for (int k = 0; k < 8; ++k) rmin[k] = fminf(rmin[k], d[k]);
    }

    // Min-reduce across 16 columns (within each 16-lane half).
#pragma unroll
    for (int off = 1; off <= 8; off <<= 1) {
#pragma unroll
        for (int k = 0; k < 8; ++k)
            rmin[k] = fminf(rmin[k], __shfl_xor(rmin[k], off, 32));
    }
    // Lanes 0-15 hold row 0..7 mins (per vgpr k), lanes 16-31 rows 8..15.
    float s = 0.0f;
#pragma unroll
    for (int k = 0; k < 8; ++k) s += rmin[k];
    s += __shfl_xor(s, 16, 32);            // sum of all 16 row mins (of -2x.y+|y|^2)

    // Add the row-constant |x_i|^2 terms: sum xa.w over the 16 rows.
    float xn = xa.w;
#pragma unroll
    for (int off = 1; off <= 8; off <<= 1) xn += __shfl_xor(xn, off, 32);

    if (lane == 0) atomicAdd(&accum[batch], s + xn);
}

__global__ void finalize_kernel(const float* __restrict__ acc,
                                float* __restrict__ out) {
    if (threadIdx.x == 0 && blockIdx.x == 0) {
        float s = 0.0f;
        for (int b = 0; b < CH_B; ++b) {
            s += acc[b]        * (1.0f / (float)CH_P);
            s += acc[CH_B + b] * (1.0f / (float)CH_Q);
        }
        out[0] = s * (1.0f / (float)CH_B);
    }
}

extern "C" void kernel_launch(void* const* d_in, const int* in_sizes, int n_in,
                              void* d_out, int out_size, void* d_ws, size_t ws_size,
                              hipStream_t stream) {
    const float* points_gen = (const float*)d_in[0];
    // d_in[1] = batch_gen (contiguous equal segments -> implicit via reshape)
    const float* label      = (const float*)d_in[2];
    float*       out        = (float*)d_out;

    const int nGen = CH_B * CH_P;
    const int nGt  = CH_B * CH_Q;

    char*   ws   = (char*)d_ws;
    float4* x4   = (float4*)ws;                         // nGen float4
    float4* y4   = x4 + nGen;                           // nGt  float4
    float2* bx   = (float2*)(y4 + nGt);                 // nGen*2 float2
    float2* by   = bx + (size_t)nGen * 2;               // nGt*2  float2
    float*  acc  = (float*)(by + (size_t)nGt * 2);      // 2*B floats

    zero_kernel<<<1, 64, 0, stream>>>(acc, 2 * CH_B);
    prep_gen_kernel<<<(nGen + 255) / 256, 256, 0, stream>>>(points_gen, x4, bx, nGen);
    prep_gt_kernel<<<(nGt + 255) / 256, 256, 0, stream>>>(label, y4, by, nGt);

    // Pass 1: rows = generated (P), cols = ground truth (Q): axis-1 mins.
    {
        const int totalWaves = CH_B * (CH_P / 16);   // 4096
        chamfer_rowmin_kernel<<<totalWaves / 8, 256, 0, stream>>>(
            x4, by, acc, CH_P, CH_Q);
    }
    // Pass 2: rows = ground truth (Q), cols = generated (P): axis-0 mins.
    {
        const int totalWaves = CH_B * (CH_Q / 16);   // 4096
        chamfer_rowmin_kernel<<<totalWaves / 8, 256, 0, stream>>>(
            y4, bx, acc + CH_B, CH_Q, CH_P);
    }

    finalize_kernel<<<1, 32, 0, stream>>>(acc, out);
}